// _CREStereoNet_83099027243079
// MI455X (gfx1250) — compile-verified
//
#include <hip/hip_runtime.h>
#include <hip/hip_bf16.h>
#include <stdint.h>

// Problem constants (from setup_inputs)
#define N_   2
#define C_   256
#define H_   128
#define W_   256
#define CG_  64      // channels per correlation group
#define NG_  4       // number of groups (== waves per block)
#define SN_  9       // SEARCH_NUM
#define OB_  3       // offsets handled per block (SN_ == 3 * OB_)
#define RX_  4
#define TW_  32      // pixel tile per block == wave32 lanes

typedef unsigned int uint32x4_ __attribute__((ext_vector_type(4)));
typedef int          int32x8_  __attribute__((ext_vector_type(8)));
typedef int          int32x4_  __attribute__((ext_vector_type(4)));

// ---------------------------------------------------------------------------
// Kernel 1: fmap2 NCHW -> NHWC (f32) into d_ws, LDS-tiled so both the global
// reads (lanes along W) and writes (lanes along C) are 128B coalesced.
// ---------------------------------------------------------------------------
__global__ __launch_bounds__(256) void transpose_nchw_to_nhwc(
    const float* __restrict__ in, float* __restrict__ out) {
  __shared__ float tile[32][33];
  const int w0 = blockIdx.x * 32;
  const int c0 = blockIdx.y * 32;
  const int nh = blockIdx.z;
  const int n = nh / H_, h = nh - n * H_;
  const int tx = threadIdx.x, ty = threadIdx.y;  // 32 x 8

#pragma unroll
  for (int i = 0; i < 32; i += 8) {
    const int c = c0 + ty + i;
    tile[ty + i][tx] =
        in[(((size_t)n * C_ + c) * H_ + h) * W_ + (w0 + tx)];
  }
  __syncthreads();
#pragma unroll
  for (int i = 0; i < 32; i += 8) {
    const int w = w0 + ty + i;
    out[(((size_t)n * H_ + h) * W_ + w) * C_ + (c0 + tx)] = tile[tx][ty + i];
  }
}

// ---------------------------------------------------------------------------
// Kernel 2: correlation. Block = 4 wave32 (x = pixel lane, y = channel group)
// covering a 32-pixel tile and 3 of the 9 search offsets (amortizes the lf
// tile). Wave 0 DMAs the fmap1 tile (C_ x TW_, 32 KB) from NCHW into LDS with
// the Tensor Data Mover; every wave reduces its own 64-channel group against
// channel-contiguous (NHWC) fmap2 corner runs with float4 gathers.
// 32 KB LDS / 4-wave block -> ~10 blocks (40 waves) per WGP.
// ---------------------------------------------------------------------------
__global__ __launch_bounds__(TW_ * NG_) void corr_kernel(
    const float* __restrict__ fmap1,    // NCHW
    const float* __restrict__ rf_nhwc,  // N H W C (from kernel 1)
    const float* __restrict__ flow,     // N 2 H W
    const float* __restrict__ extra,    // N 18 H W
    float* __restrict__ out) {          // N 36 H W
  __shared__ float lf[C_ * TW_];  // 32 KB: lf[c*TW_ + t]

  const int bx = blockIdx.x;
  const int tile = bx / OB_;
  const int ob = bx - tile * OB_;   // this block's offset triple
  const int h = blockIdx.y;
  const int n = blockIdx.z;
  const int w0 = tile * TW_;
  const int t = threadIdx.x;   // pixel lane within wave32
  const int g = threadIdx.y;   // channel group == wave id
  const int w = w0 + t;
  const size_t hw = (size_t)H_ * W_;

  // ---- stage the lf tile: wave 0 issues one TDM transfer -----------------
  const float* gsrc = fmap1 + ((size_t)n * C_ * hw) + (size_t)h * W_ + w0;
#if defined(__AMDGCN__) && __has_builtin(__builtin_amdgcn_tensor_load_to_lds)
  if (g == 0) {
    const uint64_t ga = (uint64_t)(uintptr_t)gsrc;
    const uint32_t ldsa = (uint32_t)(uintptr_t)&lf[0];  // flat[31:0] == LDS offset
    uint32x4_ g0;
    g0.x = 1u;                                   // count=1 (valid descriptor)
    g0.y = ldsa;                                 // lds_addr
    g0.z = (uint32_t)(ga & 0xFFFFFFFFu);         // global_addr[31:0]
    g0.w = (uint32_t)((ga >> 32) & 0x01FFFFFFu)  // global_addr[56:32]
           | (2u << 30);                         // type=2 ("image")
    int32x8_ g1;
    g1[0] = (int)(2u << 16);                     // data_size = 4 bytes
    g1[1] = (int)((W_ & 0xFFFF) << 16);          // tensor_dim0 = W_
    g1[2] = (int)((W_ >> 16) | ((C_ & 0xFFFF) << 16));  // | tensor_dim1 = C_
    g1[3] = (int)((C_ >> 16) | (TW_ << 16));     // tile_dim0 = TW_
    g1[4] = (int)C_;                             // tile_dim1 = C_, tile_dim2 = 0
    g1[5] = (int)(H_ * W_);                      // tensor_dim0_stride = H*W
    g1[6] = 0;                                   // strides hi / dim1_stride
    g1[7] = 0;
    const int32x4_ z4 = {0, 0, 0, 0};
#if defined(__clang_major__) && (__clang_major__ >= 23)
    const int32x8_ z8 = {0, 0, 0, 0, 0, 0, 0, 0};
    __builtin_amdgcn_tensor_load_to_lds(g0, g1, z4, z4, z8, 0);
#else
    __builtin_amdgcn_tensor_load_to_lds(g0, g1, z4, z4, 0);
#endif
  }
#else
  // Cooperative fallback: each wave loads its own channel group (coalesced).
  for (int cc = 0; cc < CG_; ++cc) {
    const int c = g * CG_ + cc;
    lf[c * TW_ + t] = gsrc[(size_t)c * hw + t];
  }
#endif

  // ---- per-pixel coordinates, overlapped with the TDM transfer -----------
  const size_t phw = (size_t)h * W_ + w;
  const float flx = flow[(size_t)n * 2 * hw + phw];        // offset-independent
  const float fly = flow[((size_t)n * 2 + 1) * hw + phw];
  float ex[OB_], ey[OB_];
#pragma unroll
  for (int k = 0; k < OB_; ++k) {
    const int o = ob * OB_ + k;
    ex[k] = extra[((size_t)n * (2 * SN_) + 2 * o) * hw + phw];
    ey[k] = extra[((size_t)n * (2 * SN_) + 2 * o + 1) * hw + phw];
  }

#if defined(__AMDGCN__) && __has_builtin(__builtin_amdgcn_s_wait_tensorcnt)
  if (g == 0) {
    __builtin_amdgcn_s_wait_tensorcnt(0);
    asm volatile("" ::: "memory");
  }
#endif
  __syncthreads();

  const float* lfg = &lf[(g * CG_) * TW_ + t];
  const float inv = 1.0f / (float)CG_;

#pragma unroll
  for (int k = 0; k < OB_; ++k) {
    const int o = ob * OB_ + k;
    const float cx = (float)w + flx + (float)(o - RX_) + ex[k];
    const float cy = (float)h + fly + ey[k];

    const float x0f = floorf(cx), y0f = floorf(cy);
    const int x0 = (int)x0f, y0 = (int)y0f;
    const float fx = cx - x0f, fy = cy - y0f;

    // corner order: (x0,y0) (x0,y1) (x1,y0) (x1,y1) == wa wb wc wd
    float wgt[4] = {(1.f - fx) * (1.f - fy), (1.f - fx) * fy,
                    fx * (1.f - fy), fx * fy};
    const int qx[4] = {x0, x0, x0 + 1, x0 + 1};
    const int qy[4] = {y0, y0 + 1, y0, y0 + 1};
    const float* base[4];
#pragma unroll
    for (int j = 0; j < 4; ++j) {
      // zero-padding: any corner outside [0,W)x[0,H) contributes 0
      if (!((unsigned)qx[j] < (unsigned)W_ && (unsigned)qy[j] < (unsigned)H_))
        wgt[j] = 0.f;
      const int cqx = qx[j] < 0 ? 0 : (qx[j] > W_ - 1 ? W_ - 1 : qx[j]);
      const int cqy = qy[j] < 0 ? 0 : (qy[j] > H_ - 1 ? H_ - 1 : qy[j]);
      base[j] = rf_nhwc + (((size_t)n * H_ + cqy) * W_ + cqx) * C_ + g * CG_;
      __builtin_prefetch(base[j], 0, 3);
      __builtin_prefetch(base[j] + 32, 0, 3);
    }

    // ---- 64-channel dot products for this wave's group: 4 corners --------
    float a0 = 0.f, a1 = 0.f, a2 = 0.f, a3 = 0.f;
#pragma unroll 8
    for (int cc = 0; cc < CG_; cc += 4) {
      const float l0 = lfg[(cc + 0) * TW_];
      const float l1 = lfg[(cc + 1) * TW_];
      const float l2 = lfg[(cc + 2) * TW_];
      const float l3 = lfg[(cc + 3) * TW_];
      const float4 r0 = *(const float4*)(base[0] + cc);
      const float4 r1 = *(const float4*)(base[1] + cc);
      const float4 r2 = *(const float4*)(base[2] + cc);
      const float4 r3 = *(const float4*)(base[3] + cc);
      a0 += l0 * r0.x + l1 * r0.y + l2 * r0.z + l3 * r0.w;
      a1 += l0 * r1.x + l1 * r1.y + l2 * r1.z + l3 * r1.w;
      a2 += l0 * r2.x + l1 * r2.y + l2 * r2.z + l3 * r2.w;
      a3 += l0 * r3.x + l1 * r3.y + l2 * r3.z + l3 * r3.w;
    }

    const float v = wgt[0] * a0 + wgt[1] * a1 + wgt[2] * a2 + wgt[3] * a3;
    out[(((size_t)n * (NG_ * SN_) + g * SN_ + o) * H_ + h) * W_ + w] = v * inv;
  }
}

// ---------------------------------------------------------------------------
extern "C" void kernel_launch(void* const* d_in, const int* in_sizes, int n_in,
                              void* d_out, int out_size, void* d_ws,
                              size_t ws_size, hipStream_t stream) {
  (void)in_sizes; (void)n_in; (void)out_size; (void)ws_size;
  const float* fmap1 = (const float*)d_in[0];
  const float* fmap2 = (const float*)d_in[1];
  const float* flow  = (const float*)d_in[2];
  const float* extra = (const float*)d_in[3];
  float* out = (float*)d_out;
  float* rf_nhwc = (float*)d_ws;  // needs N_*H_*W_*C_*4 = 64 MB of scratch

  dim3 tb(32, 8);
  dim3 tg(W_ / 32, C_ / 32, N_ * H_);
  transpose_nchw_to_nhwc<<<tg, tb, 0, stream>>>(fmap2, rf_nhwc);

  dim3 cg((W_ / TW_) * OB_, H_, N_);
  corr_kernel<<<cg, dim3(TW_, NG_), 0, stream>>>(fmap1, rf_nhwc, flow, extra,
                                                 out);
}